// LengthRegulator_23605140259248
// MI455X (gfx1250) — compile-verified
//
#include <hip/hip_runtime.h>
#include <hip/hip_bf16.h>

#define B_   32
#define S_   1024
#define D_   384
#define T_   8192
#define ROW_BYTES (D_ * 4)          // 1536 bytes per row
#define WAVES_PER_BLOCK 8

// ---------------------------------------------------------------------------
// Kernel 1: per-batch inclusive cumsum of duration (one wave32 per batch).
// Writes csum[b][S_] (int32) + mel_len (int32 to ws, int64 to d_out tail).
// ---------------------------------------------------------------------------
__global__ void lr_scan_kernel(const int* __restrict__ dur,
                               int* __restrict__ csum,
                               int* __restrict__ mel32,
                               long long* __restrict__ mel64) {
    const int b    = blockIdx.x;
    const int lane = threadIdx.x;                 // 0..31, wave32
    const int* d   = dur + b * S_;
    const int base = lane * 32;                   // 32 elements per lane

    int sum = 0;
#pragma unroll
    for (int i = 0; i < 32; ++i) sum += d[base + i];

    // inclusive wave scan of per-lane sums
    int incl = sum;
#pragma unroll
    for (int off = 1; off < 32; off <<= 1) {
        int n = __shfl_up(incl, off, 32);
        if (lane >= off) incl += n;
    }
    const int excl  = incl - sum;
    const int total = __shfl(incl, 31, 32);

    int run = excl;
    int* c  = csum + b * S_;
#pragma unroll
    for (int i = 0; i < 32; ++i) {
        run += d[base + i];
        c[base + i] = run;
    }

    if (lane == 0) {
        mel32[b] = total;
        mel64[b] = (long long)total;
    }
}

// ---------------------------------------------------------------------------
// Kernel 2: one wave per output row (b, t).
//   t >= mel_len[b] : store 1536B of zeros (plain b128 stores)
//   else            : binary-search csum for source row s, then DMA the row
//                     global -> LDS -> global with gfx1250 async ops.
// ---------------------------------------------------------------------------
__global__ __launch_bounds__(WAVES_PER_BLOCK * 32)
void lr_expand_kernel(const float* __restrict__ x,
                      const int* __restrict__ csum,
                      const int* __restrict__ mel32,
                      float* __restrict__ out) {
    __shared__ __align__(16) char lds_buf[WAVES_PER_BLOCK * ROW_BYTES];

    const unsigned wave_in_blk = threadIdx.x >> 5;
    const unsigned lane        = threadIdx.x & 31u;
    const unsigned row         = blockIdx.x * WAVES_PER_BLOCK + wave_in_blk;
    const unsigned b           = row >> 13;       // / T_
    const unsigned t           = row & (T_ - 1);  // % T_

    const int ml = mel32[b];                      // wave-uniform

    if ((int)t >= ml) {
        // zero row: 96 float4 per row, 3 per lane
        float4 z = make_float4(0.f, 0.f, 0.f, 0.f);
        float4* dst = (float4*)(out + ((size_t)b * T_ + t) * D_);
        dst[lane]      = z;
        dst[lane + 32] = z;
        dst[lane + 64] = z;
        return;                                   // uniform branch: EXEC stays full
    }

    // searchsorted(csum, t, side='right'): first s with csum[s] > t
    const int* c = csum + b * S_;
    int lo = 0, hi = S_;
#pragma unroll
    for (int it = 0; it < 10; ++it) {             // ceil(log2(1024)) = 10
        int mid = (lo + hi) >> 1;
        if (c[mid] <= (int)t) lo = mid + 1; else hi = mid;
    }
    const unsigned s = (unsigned)lo;              // < S_ because t < mel_len

    // byte offsets (all < 2^31): per-lane VGPR offset + SGPR base pointer (GVS)
    const unsigned src_off = (b * S_ + s) * (unsigned)ROW_BYTES + lane * 16u;
    const unsigned dst_off = (b * T_ + t) * (unsigned)ROW_BYTES + lane * 16u;
    const unsigned lds_a   = (unsigned)(unsigned long long)(const void*)
                                 (&lds_buf[wave_in_blk * ROW_BYTES]) + lane * 16u;

    // 3 x 512B async chunks; imm offset advances both LDS and global address
    asm volatile(
        "global_load_async_to_lds_b128 %0, %1, %2 offset:0\n\t"
        "global_load_async_to_lds_b128 %0, %1, %2 offset:512\n\t"
        "global_load_async_to_lds_b128 %0, %1, %2 offset:1024"
        :: "v"(lds_a), "v"(src_off), "s"(x) : "memory");

    asm volatile("s_wait_asynccnt 0" ::: "memory");   // LDS writes visible to async stores

    asm volatile(
        "global_store_async_from_lds_b128 %0, %1, %2 offset:0\n\t"
        "global_store_async_from_lds_b128 %0, %1, %2 offset:512\n\t"
        "global_store_async_from_lds_b128 %0, %1, %2 offset:1024"
        :: "v"(dst_off), "v"(lds_a), "s"(out) : "memory");
    // s_endpgm performs an implicit wait-idle, draining outstanding async stores
}

extern "C" void kernel_launch(void* const* d_in, const int* in_sizes, int n_in,
                              void* d_out, int out_size, void* d_ws, size_t ws_size,
                              hipStream_t stream) {
    const float* x   = (const float*)d_in[0];
    const int*   dur = (const int*)d_in[1];
    // d_in[2] = max_mel_len scalar (fixed at 8192 for these shapes)

    float* out = (float*)d_out;

    int* csum  = (int*)d_ws;                 // B_*S_ int32 = 128 KB
    int* mel32 = csum + (size_t)B_ * S_;     // B_ int32
    long long* mel64 =
        (long long*)(out + (size_t)B_ * T_ * D_);   // int64 tail of d_out

    lr_scan_kernel<<<B_, 32, 0, stream>>>(dur, csum, mel32, mel64);

    const int rows   = B_ * T_;              // 262144 output rows
    const int blocks = rows / WAVES_PER_BLOCK;
    lr_expand_kernel<<<blocks, WAVES_PER_BLOCK * 32, 0, stream>>>(x, csum, mel32, out);
}